// GuardNet_28587302322289
// MI455X (gfx1250) — compile-verified
//
#include <hip/hip_runtime.h>

#define NNODE 50000
#define FDIM  128      // feature dim for both attention rounds (DIN == DH == 128)
#define DINC  128
#define DHC   128
#define DOUTC 64

typedef __attribute__((ext_vector_type(2))) float v2f;
typedef __attribute__((ext_vector_type(8))) float v8f;

// ---------------- generic fill ----------------
__global__ void k_fill(float* __restrict__ p, int n, float v) {
  int i = blockIdx.x * blockDim.x + threadIdx.x;
  if (i < n) p[i] = v;
}

// ---------------- row-normalize: fhat = x / max(||x||,1e-12) ----------------
// one wave32 per node, 128 floats per row -> float4 per lane
__global__ void k_normalize(const float* __restrict__ x, float* __restrict__ fhat, int n) {
  int wid  = (blockIdx.x * blockDim.x + threadIdx.x) >> 5;
  int lane = threadIdx.x & 31;
  if (wid >= n) return;
  const float4* xr = (const float4*)(x + (size_t)wid * FDIM);
  float4 a = xr[lane];
  float ss = a.x*a.x + a.y*a.y + a.z*a.z + a.w*a.w;
#pragma unroll
  for (int off = 16; off; off >>= 1) ss += __shfl_xor(ss, off, 32);
  float inv = 1.0f / fmaxf(sqrtf(ss), 1e-12f);
  float4 o; o.x = a.x*inv; o.y = a.y*inv; o.z = a.z*inv; o.w = a.w*inv;
  ((float4*)(fhat + (size_t)wid * FDIM))[lane] = o;
}

// ---------------- per-edge cosine sim + row L1 sum + degree count ----------------
// one wave32 per edge
__global__ void k_edge_sim(const float* __restrict__ fhat,
                           const int* __restrict__ rowi, const int* __restrict__ coli,
                           float* __restrict__ sim, float* __restrict__ rs,
                           float* __restrict__ degc, int E) {
  int wid  = (blockIdx.x * blockDim.x + threadIdx.x) >> 5;
  int lane = threadIdx.x & 31;
  if (wid >= E) return;
  int r = rowi[wid], c = coli[wid];
  float4 a = ((const float4*)(fhat + (size_t)r * FDIM))[lane];
  float4 b = ((const float4*)(fhat + (size_t)c * FDIM))[lane];
  float s = a.x*b.x + a.y*b.y + a.z*b.z + a.w*b.w;
#pragma unroll
  for (int off = 16; off; off >>= 1) s += __shfl_xor(s, off, 32);
  if (lane == 0) {
    if (s < 0.1f || r == c) s = 0.0f;          // threshold + drop diagonal
    sim[wid] = s;
    if (s != 0.0f) {
      atomicAdd(&rs[r], fabsf(s));
      atomicAdd(&degc[r], 1.0f);
    }
  }
}

// ---------------- per-node: w_self = exp(1/(deg+1)), init weighted-degree ----------------
__global__ void k_node_wself(const float* __restrict__ degc, float* __restrict__ wself,
                             float* __restrict__ degw, int n) {
  int i = blockIdx.x * blockDim.x + threadIdx.x;
  if (i >= n) return;
  float ws = __expf(1.0f / (degc[i] + 1.0f));
  wself[i] = ws;
  degw[i]  = ws;                                // deg = segsum(w_edge, col) + w_self
}

// ---------------- per-edge: attn -> exp reweight; accumulate weighted degree ----------------
__global__ void k_edge_w(const float* __restrict__ sim,
                         const int* __restrict__ rowi, const int* __restrict__ coli,
                         const float* __restrict__ rs, float* __restrict__ wedge,
                         float* __restrict__ degw, int E) {
  int i = blockIdx.x * blockDim.x + threadIdx.x;
  if (i >= E) return;
  float s = sim[i];
  int r = rowi[i];
  float denom = rs[r];
  if (denom == 0.0f) denom = 1.0f;
  float attn = s / denom;
  float w = (attn > 0.0f) ? __expf(attn) : 0.0f;
  wedge[i] = w;
  if (w != 0.0f) atomicAdd(&degw[coli[i]], w);
}

// ---------------- per-node: dinv = rsqrt(deg) ----------------
__global__ void k_node_dinv(const float* __restrict__ degw, float* __restrict__ dinv, int n) {
  int i = blockIdx.x * blockDim.x + threadIdx.x;
  if (i >= n) return;
  float d = degw[i];
  dinv[i] = (d > 0.0f) ? rsqrtf(d) : 0.0f;
}

// ---------------- fp32 WMMA GEMM: H[n, NT*16] = X[n,k] @ W[k, NT*16] ----------------
// One wave32 per 16-row M-tile; the wave sweeps ALL NT column tiles with NT
// independent accumulators, so consecutive v_wmma_f32_16x16x4_f32 target
// different D registers (no RAW chain) and the A fragment is reused NT times.
// fp32 A 16x4 layout:  lane&15 = M row, lane>>4 selects K pair (v0=K, v1=K+1)
// fp32 B 4x16 layout:  lane&15 = N col, lane>>4 selects K pair (v0=K, v1=K+1)
// fp32 C/D 16x16:      VGPR v, lane l -> row = v + 8*(l>>4), col = l&15
template<int NT>
__global__ void k_gemm_wmma(const float* __restrict__ X, const float* __restrict__ W,
                            float* __restrict__ H, int n, int k) {
  const int dout = NT * 16;
  int wave = (int)((blockIdx.x * blockDim.x + threadIdx.x) >> 5);
  int lane = threadIdx.x & 31;
  int tm = wave;
  if (tm * 16 >= n) return;                     // wave-uniform: EXEC stays all-1s
  int mrow = tm * 16 + (lane & 15);
  int ncol = lane & 15;
  int kh   = (lane >> 4) * 2;                   // 0 or 2
  const float* xr = X + (size_t)mrow * k;
  __builtin_prefetch(xr, 0, 3);                 // global_prefetch_b8: kick the stream

  v8f c[NT];
#pragma unroll
  for (int nt = 0; nt < NT; ++nt) c[nt] = (v8f){};

#pragma unroll 2
  for (int kk = 0; kk < k; kk += 4) {
    v2f a;
    a.x = xr[kk + kh + 0];
    a.y = xr[kk + kh + 1];
    const float* w0 = W + (size_t)(kk + kh + 0) * dout + ncol;
    const float* w1 = W + (size_t)(kk + kh + 1) * dout + ncol;
#pragma unroll
    for (int nt = 0; nt < NT; ++nt) {
      v2f b;
      b.x = w0[nt * 16];
      b.y = w1[nt * 16];
      c[nt] = __builtin_amdgcn_wmma_f32_16x16x4_f32(false, a, false, b,
                                                    (short)0, c[nt], false, false);
    }
  }

  int rbase = tm * 16 + (lane >> 4) * 8;
#pragma unroll
  for (int nt = 0; nt < NT; ++nt) {
    float* hp = H + (size_t)rbase * dout + nt * 16 + ncol;
#pragma unroll
    for (int v = 0; v < 8; ++v) {
      hp[(size_t)v * dout] = c[nt][v];
    }
  }
}

// ---------------- edge scatter: acc[col] += h[row] * dinv[row]*w*dinv[col] ----------------
// one wave32 per edge
__global__ void k_scatter(const float* __restrict__ h,
                          const int* __restrict__ rowi, const int* __restrict__ coli,
                          const float* __restrict__ wedge, const float* __restrict__ dinv,
                          float* __restrict__ acc, int E, int d) {
  int wid  = (blockIdx.x * blockDim.x + threadIdx.x) >> 5;
  int lane = threadIdx.x & 31;
  if (wid >= E) return;
  float w = wedge[wid];
  if (w == 0.0f) return;
  int r = rowi[wid], c = coli[wid];
  float nw = dinv[r] * w * dinv[c];
  const float* hr = h + (size_t)r * d;
  float* ac = acc + (size_t)c * d;
  for (int j = lane; j < d; j += 32)
    atomicAdd(&ac[j], hr[j] * nw);
}

// ---------------- out = acc + h * (w_self*dinv^2) + bias, optional ReLU ----------------
__global__ void k_self_bias(const float* __restrict__ h, const float* __restrict__ acc,
                            const float* __restrict__ wself, const float* __restrict__ dinv,
                            const float* __restrict__ bias, float* __restrict__ out,
                            int n, int d, int relu) {
  int i = blockIdx.x * blockDim.x + threadIdx.x;
  if (i >= (int)(n * d)) return;
  int node = i / d, j = i - node * d;
  float di = dinv[node];
  float v = acc[i] + h[i] * (wself[node] * di * di) + bias[j];
  if (relu) v = fmaxf(v, 0.0f);
  out[i] = v;
}

// ---------------- row log_softmax over d=64 (one wave per row, 2 elems/lane) ----------------
__global__ void k_logsoftmax(float* __restrict__ out, int n) {
  int wid  = (blockIdx.x * blockDim.x + threadIdx.x) >> 5;
  int lane = threadIdx.x & 31;
  if (wid >= n) return;
  float* row = out + (size_t)wid * DOUTC;
  float v0 = row[lane], v1 = row[lane + 32];
  float m = fmaxf(v0, v1);
#pragma unroll
  for (int off = 16; off; off >>= 1) m = fmaxf(m, __shfl_xor(m, off, 32));
  float s = __expf(v0 - m) + __expf(v1 - m);
#pragma unroll
  for (int off = 16; off; off >>= 1) s += __shfl_xor(s, off, 32);
  float lse = m + __logf(s);
  row[lane]      = v0 - lse;
  row[lane + 32] = v1 - lse;
}

extern "C" void kernel_launch(void* const* d_in, const int* in_sizes, int n_in,
                              void* d_out, int out_size, void* d_ws, size_t ws_size,
                              hipStream_t stream) {
  const float* data = (const float*)d_in[0];
  const int*   eidx = (const int*)d_in[1];     // [2, E] int32
  const float* W1   = (const float*)d_in[2];
  const float* b1   = (const float*)d_in[3];
  const float* W2   = (const float*)d_in[4];
  const float* b2   = (const float*)d_in[5];
  float* out = (float*)d_out;

  const int E = in_sizes[1] / 2;
  const int* rowi = eidx;
  const int* coli = eidx + E;

  float* ws = (float*)d_ws;
  size_t off = 0;
  float* bufA  = ws + off; off += (size_t)NNODE * FDIM;   // fhat1 -> acc1 -> h2
  float* bufB  = ws + off; off += (size_t)NNODE * FDIM;   // h1 -> fhat2
  float* bufC  = ws + off; off += (size_t)NNODE * FDIM;   // x1 (relu output)
  float* sim   = ws + off; off += (size_t)E;
  float* wedge = ws + off; off += (size_t)E;
  float* rs    = ws + off; off += NNODE;
  float* degc  = ws + off; off += NNODE;
  float* wself = ws + off; off += NNODE;
  float* degw  = ws + off; off += NNODE;
  float* dinv  = ws + off; off += NNODE;

  const int TB = 256;                                    // 8 wave32s per block
  const int MTILES = NNODE / 16;                         // 3125, exact
  #define GRID1(n)  (((n) + TB - 1) / TB)                // 1 thread / elem
  #define GRIDW(n)  (((n) + 7) / 8)                      // 1 wave / elem

  // ================= layer 1 =================
  k_fill<<<GRID1(NNODE), TB, 0, stream>>>(rs, NNODE, 0.0f);
  k_fill<<<GRID1(NNODE), TB, 0, stream>>>(degc, NNODE, 0.0f);
  k_normalize<<<GRIDW(NNODE), TB, 0, stream>>>(data, bufA, NNODE);
  k_edge_sim<<<GRIDW(E), TB, 0, stream>>>(bufA, rowi, coli, sim, rs, degc, E);
  k_node_wself<<<GRID1(NNODE), TB, 0, stream>>>(degc, wself, degw, NNODE);
  k_edge_w<<<GRID1(E), TB, 0, stream>>>(sim, rowi, coli, rs, wedge, degw, E);
  k_node_dinv<<<GRID1(NNODE), TB, 0, stream>>>(degw, dinv, NNODE);

  k_gemm_wmma<DHC/16><<<GRIDW(MTILES), TB, 0, stream>>>(data, W1, bufB, NNODE, DINC);

  k_fill<<<GRID1(NNODE * DHC), TB, 0, stream>>>(bufA, NNODE * DHC, 0.0f);
  k_scatter<<<GRIDW(E), TB, 0, stream>>>(bufB, rowi, coli, wedge, dinv, bufA, E, DHC);
  k_self_bias<<<GRID1(NNODE * DHC), TB, 0, stream>>>(bufB, bufA, wself, dinv, b1, bufC,
                                                     NNODE, DHC, 1);

  // ================= layer 2 =================
  k_fill<<<GRID1(NNODE), TB, 0, stream>>>(rs, NNODE, 0.0f);
  k_fill<<<GRID1(NNODE), TB, 0, stream>>>(degc, NNODE, 0.0f);
  k_normalize<<<GRIDW(NNODE), TB, 0, stream>>>(bufC, bufB, NNODE);
  k_edge_sim<<<GRIDW(E), TB, 0, stream>>>(bufB, rowi, coli, sim, rs, degc, E);
  k_node_wself<<<GRID1(NNODE), TB, 0, stream>>>(degc, wself, degw, NNODE);
  k_edge_w<<<GRID1(E), TB, 0, stream>>>(sim, rowi, coli, rs, wedge, degw, E);
  k_node_dinv<<<GRID1(NNODE), TB, 0, stream>>>(degw, dinv, NNODE);

  k_gemm_wmma<DOUTC/16><<<GRIDW(MTILES), TB, 0, stream>>>(bufC, W2, bufA, NNODE, DHC);

  k_fill<<<GRID1(NNODE * DOUTC), TB, 0, stream>>>(out, NNODE * DOUTC, 0.0f);
  k_scatter<<<GRIDW(E), TB, 0, stream>>>(bufA, rowi, coli, wedge, dinv, out, E, DOUTC);
  k_self_bias<<<GRID1(NNODE * DOUTC), TB, 0, stream>>>(bufA, out, wself, dinv, b2, out,
                                                       NNODE, DOUTC, 0);
  k_logsoftmax<<<GRIDW(NNODE), TB, 0, stream>>>(out, NNODE);

  #undef GRID1
  #undef GRIDW
}